// DUDCLoss_1382979469646
// MI455X (gfx1250) — compile-verified
//
#include <hip/hip_runtime.h>

// DUDC loss for MI455X (gfx1250, wave32).
// Kernel 1: one 256-thread block per batch row (B=4096 blocks).
//   - loads the 1024-wide row of out1/out2 once (float4 per thread)
//   - loss_multi: sigmoid xent, accumulated locally + LDS tree reduce
//   - loss_single: shifted exps e1/e2, per-positive denominators D1_j/D2_j,
//     shared full-row sums U12[8]/U21[8] (16 accumulators) + tiny positive
//     corrections; the 16x256 partial matrix is reduced with chained
//     V_WMMA_F32_16X16X4_F32 (ones-matrix trick, exact f32).
// Kernel 2: reduces 4096 row partials and applies para mixing.

#define B_SZ    4096
#define C_SZ    1024
#define K_SZ    8
#define EPS     1e-5f
#define THREADS 256

typedef float v2f __attribute__((ext_vector_type(2)));
typedef float v8f __attribute__((ext_vector_type(8)));

__device__ __forceinline__ float block_reduce_sum(float v, float* red) {
    const int tid = threadIdx.x;
    red[tid] = v;
    __syncthreads();
    #pragma unroll
    for (int s = THREADS / 2; s > 0; s >>= 1) {
        if (tid < s) red[tid] += red[tid + s];
        __syncthreads();
    }
    float r = red[0];
    __syncthreads();
    return r;
}

__device__ __forceinline__ float block_reduce_max(float v, float* red) {
    const int tid = threadIdx.x;
    red[tid] = v;
    __syncthreads();
    #pragma unroll
    for (int s = THREADS / 2; s > 0; s >>= 1) {
        if (tid < s) red[tid] = fmaxf(red[tid], red[tid + s]);
        __syncthreads();
    }
    float r = red[0];
    __syncthreads();
    return r;
}

__global__ __launch_bounds__(THREADS) void dudc_row_kernel(
    const float* __restrict__ out1, const float* __restrict__ out2,
    const int* __restrict__ pos_idx, float* __restrict__ ws) {

    __shared__ float sE1[C_SZ];          // exp(out1 - M1)
    __shared__ float sE2[C_SZ];          // exp(out2 - M2)
    __shared__ float sU[16 * 257];       // 16 accumulators x 256 threads (padded rows)
    __shared__ float sRed[THREADS];
    __shared__ float sFin[16];
    __shared__ int   sPos[K_SZ];

    const int tid = threadIdx.x;
    const int row = blockIdx.x;
    const float* __restrict__ r1 = out1 + (size_t)row * C_SZ;
    const float* __restrict__ r2 = out2 + (size_t)row * C_SZ;

    // CDNA5 global_prefetch_b8 (speculative, RT)
    __builtin_prefetch(r1 + tid * 4, 0, 1);
    __builtin_prefetch(r2 + tid * 4, 0, 1);

    const float4 x1 = reinterpret_cast<const float4*>(r1)[tid];
    const float4 x2 = reinterpret_cast<const float4*>(r2)[tid];
    if (tid < K_SZ) sPos[tid] = pos_idx[row * K_SZ + tid];

    // ---- row maxima (shared shift; exact ratios are shift-invariant) ----
    const float m1 = fmaxf(fmaxf(x1.x, x1.y), fmaxf(x1.z, x1.w));
    const float m2 = fmaxf(fmaxf(x2.x, x2.y), fmaxf(x2.z, x2.w));
    const float M1 = block_reduce_max(m1, sRed);
    const float M2 = block_reduce_max(m2, sRed);

    // ---- shifted exps + sigmoid xent accumulation ----
    float xv1[4] = {x1.x, x1.y, x1.z, x1.w};
    float xv2[4] = {x2.x, x2.y, x2.z, x2.w};
    float e1v[4], e2v[4];
    float s1l = 0.f, s2l = 0.f, lm12l = 0.f, lm21l = 0.f;
    #pragma unroll
    for (int c = 0; c < 4; ++c) {
        e1v[c] = __expf(xv1[c] - M1);
        e2v[c] = __expf(xv2[c] - M2);
        s1l += e1v[c];
        s2l += e2v[c];
        sE1[tid * 4 + c] = e1v[c];
        sE2[tid * 4 + c] = e2v[c];
        const float sg1 = 1.0f / (1.0f + __expf(-xv1[c]));
        const float sg2 = 1.0f / (1.0f + __expf(-xv2[c]));
        lm12l -= sg1 * __logf(sg2 + EPS);
        lm21l -= sg2 * __logf(sg1 + EPS);
    }

    const float S1   = block_reduce_sum(s1l, sRed);
    const float S2   = block_reduce_sum(s2l, sRed);
    const float LM12 = block_reduce_sum(lm12l, sRed);
    const float LM21 = block_reduce_sum(lm21l, sRed);
    // reductions above also fence sE1/sE2/sPos visibility

    // ---- per-positive denominators (redundant per thread; LDS broadcast reads) ----
    float e1p[K_SZ], e2p[K_SZ];
    float sumP1 = 0.f, sumP2 = 0.f;
    #pragma unroll
    for (int j = 0; j < K_SZ; ++j) {
        const int p = sPos[j];
        e1p[j] = sE1[p];
        e2p[j] = sE2[p];
        sumP1 += e1p[j];
        sumP2 += e2p[j];
    }
    const float Sn1 = S1 - sumP1;   // sum of exps over negatives
    const float Sn2 = S2 - sumP2;
    float invD1[K_SZ], invD2[K_SZ];
    #pragma unroll
    for (int j = 0; j < K_SZ; ++j) {
        invD1[j] = 1.0f / (Sn1 + e1p[j]);
        invD2[j] = 1.0f / (Sn2 + e2p[j]);
    }

    // ---- hot loop: 16 logs per element (K x 2 directions), full-row sums ----
    float U12[K_SZ] = {0.f, 0.f, 0.f, 0.f, 0.f, 0.f, 0.f, 0.f};
    float U21[K_SZ] = {0.f, 0.f, 0.f, 0.f, 0.f, 0.f, 0.f, 0.f};
    #pragma unroll
    for (int c = 0; c < 4; ++c) {
        const float e1c = e1v[c];
        const float e2c = e2v[c];
        #pragma unroll
        for (int j = 0; j < K_SZ; ++j) {
            U12[j] = fmaf(e1c, __logf(fmaf(e2c, invD2[j], EPS)), U12[j]);
            U21[j] = fmaf(e2c, __logf(fmaf(e1c, invD1[j], EPS)), U21[j]);
        }
    }
    #pragma unroll
    for (int j = 0; j < K_SZ; ++j) {
        sU[j * 257 + tid]       = U12[j];
        sU[(8 + j) * 257 + tid] = U21[j];
    }
    __syncthreads();

    // ---- wave 0: reduce 16 accumulators x 256 partials via ones-matrix WMMA ----
    // D[m][n] = sum_k A[m][k]*B[k][n] with A==1 -> column sums, exact f32.
    if (tid < 32) {                         // full wave32 -> EXEC all ones
        const int n    = tid & 15;          // accumulator index (N striping)
        const int koff = (tid < 16) ? 0 : 2;
        v2f a;
        a[0] = 1.0f;
        a[1] = 1.0f;
        v8f acc = {};
        const float* bp = &sU[n * 257 + koff];
        for (int k0 = 0; k0 < THREADS; k0 += 4) {
            v2f b;
            b[0] = bp[k0];
            b[1] = bp[k0 + 1];
            acc = __builtin_amdgcn_wmma_f32_16x16x4_f32(
                /*neg_a=*/false, a, /*neg_b=*/false, b,
                /*c_mod=*/(short)0, acc, /*reuse_a=*/false, /*reuse_b=*/false);
        }
        const float Utot = acc[0];          // M=0 row, N = lane (lanes 0-15)

        if (tid < 16) {
            float x;
            if (n < 8) {                    // direction 1->2, j = n
                const int j = n;
                float corr = 0.f;
                #pragma unroll
                for (int jp = 0; jp < K_SZ; ++jp)
                    if (jp != j) corr += e1p[jp] * __logf(fmaf(e2p[jp], invD2[j], EPS));
                x = -invD1[j] * (Utot - corr);
            } else {                        // direction 2->1, j = n-8
                const int j = n - 8;
                float corr = 0.f;
                #pragma unroll
                for (int jp = 0; jp < K_SZ; ++jp)
                    if (jp != j) corr += e2p[jp] * __logf(fmaf(e1p[jp], invD1[j], EPS));
                x = -invD2[j] * (Utot - corr);
            }
            sFin[n] = x;
        }
    }
    __syncthreads();

    if (tid == 0) {
        float rs = 0.f;
        #pragma unroll
        for (int n = 0; n < 16; ++n) rs += sFin[n];
        ws[2 * row]     = rs * (1.0f / (float)K_SZ);  // mean over K of (x12+x21)
        ws[2 * row + 1] = LM12 + LM21;                // multi part for this row
    }
}

__global__ __launch_bounds__(THREADS) void dudc_final_kernel(
    const float* __restrict__ ws, const float* __restrict__ para,
    float* __restrict__ out) {
    __shared__ float red[THREADS];
    float ss = 0.f, sm = 0.f;
    for (int r = threadIdx.x; r < B_SZ; r += THREADS) {
        ss += ws[2 * r];
        sm += ws[2 * r + 1];
    }
    const float S_single = block_reduce_sum(ss, red);
    const float S_multi  = block_reduce_sum(sm, red);
    if (threadIdx.x == 0) {
        const float p = para[0];
        out[0] = p * (S_multi * (1.0f / (float)B_SZ)) +
                 (1.0f - p) * (S_single * (1.0f / (float)B_SZ));
    }
}

extern "C" void kernel_launch(void* const* d_in, const int* in_sizes, int n_in,
                              void* d_out, int out_size, void* d_ws, size_t ws_size,
                              hipStream_t stream) {
    (void)in_sizes; (void)n_in; (void)out_size; (void)ws_size;
    const float* out1    = (const float*)d_in[0];
    const float* out2    = (const float*)d_in[1];
    const float* para    = (const float*)d_in[2];
    // d_in[3] = target (redundant with pos_idx; not read -> saves 16 MB of traffic)
    const int*   pos_idx = (const int*)d_in[4];
    float* ws  = (float*)d_ws;    // needs B_SZ*2 floats = 32 KB
    float* out = (float*)d_out;

    dudc_row_kernel<<<B_SZ, THREADS, 0, stream>>>(out1, out2, pos_idx, ws);
    dudc_final_kernel<<<1, THREADS, 0, stream>>>(ws, para, out);
}